// ArcFace_s_26336739459524
// MI455X (gfx1250) — compile-verified
//
#include <hip/hip_runtime.h>
#include <hip/hip_bf16.h>

typedef __attribute__((ext_vector_type(4))) unsigned int u32x4;
typedef __attribute__((ext_vector_type(8))) int          i32x8;
typedef __attribute__((ext_vector_type(4))) int          i32x4;
typedef __attribute__((ext_vector_type(4))) float        v4f;

#define ARC_S      64.0f
#define ARC_MARGIN 0.5f
#define CHUNK      4096   // floats per block = 16 KB LDS tile

// ---------------------------------------------------------------------------
// Streaming kernel: TDM (tensor_load_to_lds) pulls a 16KB tile into LDS,
// then 8 waves scale by S and stream it back out with non-temporal b128
// stores. Pure bandwidth: ~205 MB total -> ~9 us at 23.3 TB/s.
// ---------------------------------------------------------------------------
__global__ __launch_bounds__(256) void arcface_scale_kernel(
    const float* __restrict__ in, float* __restrict__ out,
    unsigned long long total) {
  __shared__ float buf[CHUNK];

  unsigned long long base = (unsigned long long)blockIdx.x * (unsigned long long)CHUNK;
  if (base >= total) return;

  if (threadIdx.x < 32) {  // wave 0 of the block drives the TDM
    unsigned long long gaddr =
        (unsigned long long)(uintptr_t)in + base * 4ull;
    // Low 32 bits of the flat shared-memory address == LDS byte offset.
    unsigned int lds_off = __builtin_amdgcn_readfirstlane(
        (unsigned int)(uintptr_t)(&buf[0]));
    unsigned long long remll = total - base;
    unsigned int dim0 = (remll >= (unsigned long long)CHUNK)
                            ? (unsigned int)CHUNK : (unsigned int)remll;

    // ---- D# group 0 (128b): count | lds_addr | global_addr | type=2 ----
    u32x4 g0;
    g0.x = 1u;                                             // count = 1 valid descriptor
    g0.y = lds_off;                                        // lds_addr[31:0]
    g0.z = (unsigned int)(gaddr & 0xFFFFFFFFull);          // global_addr[31:0]
    g0.w = (unsigned int)((gaddr >> 32) & 0x01FFFFFFull)   // global_addr[56:32]
         | (2u << 30);                                     // type = 2 ("image")

    // ---- D# group 1 (256b): sizes / dims / strides ----
    i32x8 g1;
    g1[0] = (int)(2u << 16);                     // workgroup_mask=0, data_size=2 (4B)
    g1[1] = (int)((dim0 & 0xFFFFu) << 16);       // tensor_dim0[15:0]  @ bits 63:48
    g1[2] = (int)(((dim0 >> 16) & 0xFFFFu)       // tensor_dim0[31:16] @ bits 79:64
         |  (1u << 16));                         // tensor_dim1 = 1    @ bits 95:80(lo16)
    g1[3] = (int)(((unsigned)CHUNK) << 16);      // tile_dim0 = 4096   @ bits 127:112
    g1[4] = 1;                                   // tile_dim1 = 1, tile_dim2 = 0
    g1[5] = (int)dim0;                           // tensor_dim0_stride (unused: 1 row)
    g1[6] = 0;                                   // stride hi / dim1_stride lo
    g1[7] = 0;                                   // dim1_stride hi
    i32x4 g2 = {0, 0, 0, 0};                     // <=2D tensor: groups 2/3 zero
    i32x4 g3 = {0, 0, 0, 0};
    i32x8 g4 = {0, 0, 0, 0, 0, 0, 0, 0};         // extra group (unused, clang-23 form)

    __builtin_amdgcn_tensor_load_to_lds(g0, g1, g2, g3, g4, 0);
    __builtin_amdgcn_s_wait_tensorcnt(0);
  }
  __syncthreads();

  const v4f* lbuf = (const v4f*)buf;
#pragma unroll
  for (int k = 0; k < 4; ++k) {
    unsigned int vi = threadIdx.x + (unsigned int)k * 256u;   // float4 index in tile
    unsigned long long e = base + (unsigned long long)vi * 4ull;
    if (e + 4ull <= total) {
      v4f v = lbuf[vi] * ARC_S;
      __builtin_nontemporal_store(v, (v4f*)(out + e));        // global_store_b128, TH=NT
    } else if (e < total) {
      for (unsigned long long j = e; j < total; ++j)
        out[j] = ARC_S * buf[vi * 4u + (unsigned int)(j - e)];
    }
  }
}

// ---------------------------------------------------------------------------
// Fixup: one lane per row rewrites the single target-class element.
// ---------------------------------------------------------------------------
__global__ __launch_bounds__(256) void arcface_fixup_kernel(
    const float* __restrict__ logits, const int* __restrict__ labels,
    float* __restrict__ out, int B, long long C) {
  int r = blockIdx.x * blockDim.x + threadIdx.x;
  if (r >= B) return;
  int lab = labels[r];
  if (lab < 0) return;
  long long idx = (long long)r * C + (long long)lab;
  float x = logits[idx];
  x = fminf(fmaxf(x, -1.0f), 1.0f);
  float y = cosf(acosf(x) + ARC_MARGIN);
  out[idx] = ARC_S * y;
}

extern "C" void kernel_launch(void* const* d_in, const int* in_sizes, int n_in,
                              void* d_out, int out_size, void* d_ws, size_t ws_size,
                              hipStream_t stream) {
  const float* logits = (const float*)d_in[0];
  const int*   labels = (const int*)d_in[1];
  float*       out    = (float*)d_out;

  unsigned long long total = (unsigned long long)in_sizes[0];
  int B = in_sizes[1];
  long long C = (long long)(total / (unsigned long long)B);

  unsigned int blocks =
      (unsigned int)((total + (unsigned long long)CHUNK - 1) / (unsigned long long)CHUNK);
  arcface_scale_kernel<<<blocks, 256, 0, stream>>>(logits, out, total);
  arcface_fixup_kernel<<<(B + 255) / 256, 256, 0, stream>>>(logits, labels, out, B, C);
}